// TopK_27453430956525
// MI455X (gfx1250) — compile-verified
//
#include <hip/hip_runtime.h>

// Top-k(2048) over 128 rows x 262144 fp32, one workgroup (1024 threads = 32 waves) per row.
// Pass 1: async global->LDS staged streaming (4 tiles in flight) into a replicated
//         12-bit histogram (4 replicas to cut ds_add same-address serialization).
// Pass 2: threshold compaction into LDS, then in-LDS bitonic sort of (key, ~idx)
//         64-bit packets -> descending values, ties by lowest index (matches lax.top_k).

#define ROWS     128
#define NCOL     262144
#define KSEL     2048
#define NT       1024
#define TILE_F   (NT * 4)          // floats per tile (16 B per lane => 16 KB tile)
#define NTILES   (NCOL / TILE_F)   // 64
#define NBINS    4096
#define NREP     4                 // histogram replicas (one per 8-wave group)
#define NBUF     4                 // async staging ring depth
#define CAND_CAP 8192
#define DEF_CAP  2048

__device__ __forceinline__ unsigned int mono_key(unsigned int u) {
  // order-preserving map: float bits -> unsigned (larger float => larger key)
  return u ^ ((unsigned int)((int)u >> 31) | 0x80000000u);
}

__device__ __forceinline__ void bitonic_sort_desc(unsigned long long* a, int n, int tid) {
  for (int k = 2; k <= n; k <<= 1) {
    for (int j = k >> 1; j > 0; j >>= 1) {
      for (int i = tid; i < n; i += NT) {
        int ixj = i ^ j;
        if (ixj > i) {
          unsigned long long x = a[i], y = a[ixj];
          bool desc = ((i & k) == 0);
          bool do_swap = desc ? (x < y) : (x > y);
          if (do_swap) { a[i] = y; a[ixj] = x; }
        }
      }
      __syncthreads();
    }
  }
}

__global__ __launch_bounds__(NT, 1)
void TopK_27453430956525_kernel(const float* __restrict__ logits, int* __restrict__ out) {
  __shared__ unsigned long long bufA[CAND_CAP];       // 64 KB: staging ring / candidates
  __shared__ unsigned int       histmem[NREP * NBINS];// 64 KB: replicated histogram
  __shared__ unsigned long long bufB[DEF_CAP];        // 16 KB: definite top-k
  __shared__ unsigned int       csum[NT];             //  4 KB: chunk sums
  __shared__ unsigned int       ctrl[4];              // T1, cntAbove, defCnt, candCnt

  const int tid = threadIdx.x;
  const int row = blockIdx.x;

  float* stage = (float*)bufA;                        // NBUF tiles of TILE_F floats
  unsigned int* myhist = histmem + (tid >> 8) * NBINS; // 8 waves share one replica

  for (int i = tid; i < NREP * NBINS; i += NT) histmem[i] = 0u;
  if (tid < 4) ctrl[tid] = 0u;
  __syncthreads();

  // ---------------- Pass 1: async global->LDS staged 12-bit histogram ----------------
  const unsigned long long gbase =
      (unsigned long long)(size_t)(const void*)(logits + (size_t)row * NCOL);
  const unsigned int ldsA = (unsigned int)(size_t)(void*)stage; // LDS byte offset

  auto issue = [&](int tile) {
    const int pp = tile & (NBUF - 1);
    unsigned int       lds = ldsA + (unsigned int)(pp * (TILE_F * 4) + tid * 16);
    unsigned long long ga  = gbase + (unsigned long long)tile * (TILE_F * 4)
                                   + (unsigned long long)tid * 16ull;
    asm volatile("global_load_async_to_lds_b128 %0, %1, off"
                 :: "v"(lds), "v"(ga) : "memory");
  };

  issue(0); issue(1); issue(2);                 // prime the pipeline (NTILES >= 3)
  for (int tile = 0; tile < NTILES; ++tile) {
    const int pp = tile & (NBUF - 1);
    if (tile + 3 < NTILES) {
      issue(tile + 3);
      asm volatile("s_wait_asynccnt 0x3" ::: "memory"); // oldest (this tile) landed
    } else {
      asm volatile("s_wait_asynccnt 0x0" ::: "memory"); // drain tail
    }
    const float4 v = *(const float4*)(&stage[pp * TILE_F + tid * 4]);
    const float vals[4] = {v.x, v.y, v.z, v.w};
#pragma unroll
    for (int c = 0; c < 4; ++c) {
      unsigned int key = mono_key(__float_as_uint(vals[c]));
      atomicAdd(&myhist[key >> 20], 1u); // ds_add_u32, no return
    }
  }
  __syncthreads();

  // ---------------- Reduce replicas + suffix scan from the top bin ----------------
  {
    unsigned int chunk = 0;
#pragma unroll
    for (int q = 0; q < 4; ++q) {
      int b = 4 * tid + q;
      unsigned int tot = histmem[b] + histmem[NBINS + b] +
                         histmem[2 * NBINS + b] + histmem[3 * NBINS + b];
      histmem[b] = tot;     // replica 0 now holds per-bin totals
      chunk += tot;
    }
    csum[tid] = chunk;
  }
  __syncthreads();
  if (tid == 0) {
    unsigned int acc = 0;
    int c = NT - 1;
    for (; c > 0; --c) {
      if (acc + csum[c] >= (unsigned int)KSEL) break;
      acc += csum[c];
    }
    int b = c * 4 + 3;
    for (; b > c * 4; --b) {
      if (acc + histmem[b] >= (unsigned int)KSEL) break;
      acc += histmem[b];
    }
    ctrl[0] = (unsigned int)b; // threshold bin T1
    ctrl[1] = acc;             // count strictly above T1 (< KSEL by construction)
  }
  __syncthreads();
  const unsigned int T1       = ctrl[0];
  const unsigned int cntAbove = ctrl[1];
  __syncthreads(); // all threads have T1/cntAbove before buffers are repurposed

  // zero-fill both sort buffers (0 = smallest packet -> sorts to the tail)
  for (int i = tid; i < CAND_CAP; i += NT) bufA[i] = 0ull;
  for (int i = tid; i < DEF_CAP;  i += NT) bufB[i] = 0ull;
  __syncthreads();

  // ---------------- Pass 2: compaction (re-read mostly hits the 192 MB L2) ----------------
  const float4* src = (const float4*)(logits + (size_t)row * NCOL);
  for (int t = tid; t < NCOL / 4; t += NT) {
    const float4 v = src[t];
    const float vals[4] = {v.x, v.y, v.z, v.w};
#pragma unroll
    for (int c = 0; c < 4; ++c) {
      unsigned int key = mono_key(__float_as_uint(vals[c]));
      unsigned int bin = key >> 20;
      if (bin < T1) continue;
      unsigned int idx = (unsigned int)(t * 4 + c);
      // descending sort on this packet => value desc, ties by lowest index first
      unsigned long long pk = ((unsigned long long)key << 32) | (unsigned int)(~idx);
      if (bin > T1) {
        unsigned int p = atomicAdd(&ctrl[2], 1u);
        if (p < DEF_CAP) bufB[p] = pk;
      } else {
        unsigned int p = atomicAdd(&ctrl[3], 1u);
        if (p < CAND_CAP) bufA[p] = pk;
      }
    }
  }
  __syncthreads();

  // ---------------- Sort & emit ----------------
  bitonic_sort_desc(bufB, DEF_CAP, tid);
  bitonic_sort_desc(bufA, CAND_CAP, tid);

  int* orow = out + (size_t)row * KSEL;
  const unsigned int nDef = (cntAbove < (unsigned int)DEF_CAP) ? cntAbove : (unsigned int)DEF_CAP;
  for (int i = tid; i < KSEL; i += NT) {
    unsigned long long pk = (i < (int)nDef) ? bufB[i] : bufA[i - (int)nDef];
    orow[i] = (int)(~(unsigned int)(pk & 0xffffffffull));
  }
}

extern "C" void kernel_launch(void* const* d_in, const int* in_sizes, int n_in,
                              void* d_out, int out_size, void* d_ws, size_t ws_size,
                              hipStream_t stream) {
  const float* logits = (const float*)d_in[0]; // (1, 128, 262144) fp32
  int* out = (int*)d_out;                      // (1, 128, 2048) int32 indices
  (void)in_sizes; (void)n_in; (void)out_size; (void)d_ws; (void)ws_size;
  TopK_27453430956525_kernel<<<dim3(ROWS), dim3(NT), 0, stream>>>(logits, out);
}